// NCC_15779709846355
// MI455X (gfx1250) — compile-verified
//
#include <hip/hip_runtime.h>
#include <stdint.h>

// Problem geometry (fixed by the reference): (N=2, C=1, 160^3), 9^3 box window.
#define NB    2
#define DE    160            // cube edge
#define RAD   4              // 9//2
#define KW    9
#define TS    32             // output tile edge (x and y)
#define HS    40             // halo tile edge (TS + 2*RAD)
#define HAREA (HS * HS)      // 1600 halo cells
#define NSLOT 7              // ceil(HAREA / 256)
#define ZL    10             // z outputs per block
#define ZCH   (DE / ZL)      // 16
#define TN    (DE / TS)      // 5 tiles per dim
#define PLANE (DE * DE)
#define VOL   (DE * PLANE)
#define SZB_STR 1600         // sZB field-plane stride (floats)
#define SYB_ROW 41           // padded row stride (odd -> conflict-free col reads)
#define SYB_STR (TS * SYB_ROW)   // 1312 floats per field
#define SXB_ROW 33           // padded row stride
#define SXB_STR (TS * SXB_ROW)   // 1056 floats per field
#define INV_KVOL (1.0f / 729.0f)

// ---- CDNA5 async global->LDS path (probe-verified on this toolchain) ----
#if defined(__AMDGCN__) && __has_builtin(__builtin_amdgcn_global_load_async_to_lds_b32)
  #define HAVE_ASYNC_LDS 1
#else
  #define HAVE_ASYNC_LDS 0
#endif

#if HAVE_ASYNC_LDS
typedef __attribute__((address_space(1))) int gas_int;
typedef __attribute__((address_space(3))) int las_int;

__device__ __forceinline__ void async_g2l_b32(const float* g, float* l) {
  // Generic LDS pointer: low 32 bits are the LDS byte offset.
  gas_int* gp = (gas_int*)(uintptr_t)g;
  las_int* lp = (las_int*)(uint32_t)(uintptr_t)l;
  __builtin_amdgcn_global_load_async_to_lds_b32(gp, lp, 0, 0);
}

__device__ __forceinline__ void async_wait_all() {
#if __has_builtin(__builtin_amdgcn_s_wait_asynccnt)
  __builtin_amdgcn_s_wait_asynccnt(0);
#else
  asm volatile("s_wait_asynccnt 0" ::: "memory");
#endif
}
#endif

// Fully fused LNCC: z-box in registers (running window over planes),
// y-box and x-box as sliding windows in LDS, NCC + reduction at the end.
__global__ __launch_bounds__(256)
void ncc_fused_kernel(const float* __restrict__ pred,
                      const float* __restrict__ target,
                      double* __restrict__ acc) {
  // Region A: sZB [5][1600] (z-boxed fields); aliased by sXB [5][1056] and sRed.
  __shared__ __align__(16) float smA[5 * SZB_STR];   // 32000 B
  // Region B: sYB [5][32*41]; aliased by sTP (1600 float2 staging buffer).
  __shared__ __align__(16) float smB[5 * SYB_STR];   // 26240 B
  float2* sTP = (float2*)smB;

  const int tid = threadIdx.x;
  int b = blockIdx.x;
  const int tz = b % ZCH; b /= ZCH;
  const int tx = b % TN;  b /= TN;
  const int ty = b % TN;  b /= TN;
  const int n  = b;

  const int x0 = tx * TS - RAD;
  const int y0 = ty * TS - RAD;
  const int z0 = tz * ZL;

  const float* tBase = target + (size_t)n * VOL;
  const float* pBase = pred   + (size_t)n * VOL;

  // Per-slot halo-cell geometry (slot p owns halo cell i = tid + p*256).
  int  cOff[NSLOT];
  bool cIn[NSLOT];
#pragma unroll
  for (int p = 0; p < NSLOT; ++p) {
    const int i  = tid + p * 256;
    const int ix = i % HS, iy = i / HS;
    const int gx = x0 + ix, gy = y0 + iy;
    const bool ok = (i < HAREA) && gx >= 0 && gx < DE && gy >= 0 && gy < DE;
    cIn[p]  = ok;
    cOff[p] = ok ? (gy * DE + gx) : 0;
  }

  // Running z-window sums of the five product fields, per owned halo cell.
  float runS[NSLOT][5];
#pragma unroll
  for (int p = 0; p < NSLOT; ++p)
#pragma unroll
    for (int f = 0; f < 5; ++f) runS[p][f] = 0.0f;

  // Prologue: accumulate planes [z0-4, z0+3].
  for (int s = 0; s < 2 * RAD; ++s) {
    const int zi = z0 - RAD + s;
    if (zi < 0) continue;
    const size_t zoff = (size_t)zi * PLANE;
#pragma unroll
    for (int p = 0; p < NSLOT; ++p) {
      if (cIn[p]) {
        const float tv = tBase[zoff + cOff[p]];
        const float pv = pBase[zoff + cOff[p]];
        runS[p][0] += tv;      runS[p][1] += pv;
        runS[p][2] += tv * tv; runS[p][3] += pv * pv;
        runS[p][4] += tv * pv;
      }
    }
  }

  double part = 0.0;

  for (int zo = z0; zo < z0 + ZL; ++zo) {
    // ---- stage plane zo+4 via async DMA into LDS, then fold into runS ----
    const int za = zo + RAD;
    if (za < DE) {
      const size_t zoff = (size_t)za * PLANE;
#pragma unroll
      for (int p = 0; p < NSLOT; ++p) {
        if (cIn[p]) {
          const int i = tid + p * 256;
#if HAVE_ASYNC_LDS
          async_g2l_b32(tBase + zoff + cOff[p], &sTP[i].x);
          async_g2l_b32(pBase + zoff + cOff[p], &sTP[i].y);
#else
          sTP[i] = make_float2(tBase[zoff + cOff[p]], pBase[zoff + cOff[p]]);
#endif
          if (za + 1 < DE) {  // warm next plane (global_prefetch_b8)
            __builtin_prefetch((const void*)(tBase + zoff + cOff[p] + PLANE), 0, 1);
            __builtin_prefetch((const void*)(pBase + zoff + cOff[p] + PLANE), 0, 1);
          }
        }
      }
#if HAVE_ASYNC_LDS
      async_wait_all();                 // producer == consumer: no barrier needed
      asm volatile("" ::: "memory");
#endif
#pragma unroll
      for (int p = 0; p < NSLOT; ++p) {
        if (cIn[p]) {
          const float2 tp = sTP[tid + p * 256];
          runS[p][0] += tp.x;        runS[p][1] += tp.y;
          runS[p][2] += tp.x * tp.x; runS[p][3] += tp.y * tp.y;
          runS[p][4] += tp.x * tp.y;
        }
      }
    }

    // ---- publish z-boxed fields for this output plane ----
#pragma unroll
    for (int p = 0; p < NSLOT; ++p) {
      const int i = tid + p * 256;
      if (i < HAREA) {
#pragma unroll
        for (int f = 0; f < 5; ++f) smA[f * SZB_STR + i] = runS[p][f];
      }
    }

    // ---- evict plane zo-4 (re-read from cache; values identical -> exact) ----
    const int zs = zo - RAD;
    if (zs >= 0) {
      const size_t zoff = (size_t)zs * PLANE;
#pragma unroll
      for (int p = 0; p < NSLOT; ++p) {
        if (cIn[p]) {
          const float tv = tBase[zoff + cOff[p]];
          const float pv = pBase[zoff + cOff[p]];
          runS[p][0] -= tv;      runS[p][1] -= pv;
          runS[p][2] -= tv * tv; runS[p][3] -= pv * pv;
          runS[p][4] -= tv * pv;
        }
      }
    }
    __syncthreads();   // B1: sZB ready

    // ---- y-box: 200 column tasks (f, xi), sliding window along y ----
    for (int task = tid; task < 5 * HS; task += 256) {
      const int xi = task % HS;
      const int f  = task / HS;
      const float* zp  = &smA[f * SZB_STR];
      float*       yfp = &smB[f * SYB_STR];
      float a = 0.0f;
#pragma unroll
      for (int dy = 0; dy < KW; ++dy) a += zp[dy * HS + xi];
      yfp[xi] = a;
      for (int k = 1; k < TS; ++k) {
        a += zp[(k + 8) * HS + xi] - zp[(k - 1) * HS + xi];
        yfp[k * SYB_ROW + xi] = a;
      }
    }
    __syncthreads();   // B2: sYB ready, sZB free (sXB may overwrite it)

    // ---- x-box: 160 row tasks (f, y), sliding window along x ----
    for (int task = tid; task < 5 * TS; task += 256) {
      const int y = task % TS;
      const int f = task / TS;
      const float* yp   = &smB[f * SYB_STR + y * SYB_ROW];
      float*       xrow = &smA[f * SXB_STR + y * SXB_ROW];
      float a = 0.0f;
#pragma unroll
      for (int dx = 0; dx < KW; ++dx) a += yp[dx];
      xrow[0] = a;
      for (int k = 1; k < TS; ++k) {
        a += yp[k + 8] - yp[k - 1];
        xrow[k] = a;
      }
    }
    __syncthreads();   // B3: sXB ready, sYB free

    // ---- NCC: 4 outputs per thread ----
    const int ox  = tid & (TS - 1);
    const int oyb = (tid >> 5) * 4;
#pragma unroll
    for (int q = 0; q < 4; ++q) {
      const int oy = oyb + q;
      const float tS  = smA[0 * SXB_STR + oy * SXB_ROW + ox];
      const float pS  = smA[1 * SXB_STR + oy * SXB_ROW + ox];
      const float t2S = smA[2 * SXB_STR + oy * SXB_ROW + ox];
      const float p2S = smA[3 * SXB_STR + oy * SXB_ROW + ox];
      const float tpS = smA[4 * SXB_STR + oy * SXB_ROW + ox];
      const float t_avg = tS * INV_KVOL;
      const float p_avg = pS * INV_KVOL;
      const float cross = tpS - p_avg * tS;
      const float t_var = t2S - t_avg * tS;
      const float p_var = p2S - p_avg * pS;
      part += (double)((cross * cross) / (t_var * p_var + 1e-5f));
    }
    __syncthreads();   // B4: protect sXB(=sZB) and sTP(=sYB) for next plane
  }

  // ---- block reduction (sRed aliases region A) + one f64 atomic ----
  double* sRed = (double*)smA;
  sRed[tid] = part;
  __syncthreads();
  for (int s2 = 128; s2 > 0; s2 >>= 1) {
    if (tid < s2) sRed[tid] += sRed[tid + s2];
    __syncthreads();
  }
  if (tid == 0) atomicAdd(acc, sRed[0]);
}

__global__ void ncc_init_kernel(double* acc) { *acc = 0.0; }

__global__ void ncc_fin_kernel(const double* __restrict__ acc,
                               float* __restrict__ out) {
  out[0] = (float)(-(*acc) / (double)((size_t)NB * VOL));
}

extern "C" void kernel_launch(void* const* d_in, const int* in_sizes, int n_in,
                              void* d_out, int out_size, void* d_ws, size_t ws_size,
                              hipStream_t stream) {
  (void)in_sizes; (void)n_in; (void)out_size; (void)ws_size;
  const float* pred   = (const float*)d_in[0];
  const float* target = (const float*)d_in[1];
  double* acc = (double*)d_ws;   // 8 bytes of scratch
  float*  out = (float*)d_out;

  ncc_init_kernel<<<1, 1, 0, stream>>>(acc);
  const int nblocks = NB * TN * TN * ZCH;   // 2*5*5*16 = 800
  ncc_fused_kernel<<<nblocks, 256, 0, stream>>>(pred, target, acc);
  ncc_fin_kernel<<<1, 1, 0, stream>>>(acc, out);
}